// CantorSetAttention_66425964200587
// MI455X (gfx1250) — compile-verified
//
#include <hip/hip_runtime.h>
#include <vector>
#include <set>
#include <cstring>
#include <cmath>

// ---------------- problem constants (from setup_inputs) ----------------
#define NDEPTH 8
#define B_  4
#define L_  4096
#define D_  1024
#define KP  144            // union keys padded to multiple of 16 (and of 4)
#define NT  (KP / 16)      // 9 score N-tiles per wave
#define DC  64             // D-chunk staged through LDS
#define NCHUNK (D_ / DC)   // 16
#define WAVES 2
#define BQ  (WAVES * 16)   // 32 queries per block
// LDS pitches (floats)
#define QP  68             // Q-stage pitch (16B-aligned rows for b128 stores)
#define KPITCH 68          // K/V-stage pitch (row-major both phases)
#define CP  146            // score/coefficient pitch (even -> 8B-aligned float2)
// LDS region sizes (dwords)
#define REG1 9792          // KP*KPITCH (K-stage phase A, V-stage phase C)
#define REG2 4672          // max(BQ*QP=2176, WAVES*16*CP=4672)

typedef float v2f __attribute__((ext_vector_type(2)));
typedef float v8f __attribute__((ext_vector_type(8)));
typedef int   v4i __attribute__((vector_size(16)));   // matches async builtin pointee

#define GLOBAL_AS __attribute__((address_space(1)))
#define LDS_AS    __attribute__((address_space(3)))

#if __has_builtin(__builtin_amdgcn_global_load_async_to_lds_b128)
#define HAVE_ASYNC 1
#else
#define HAVE_ASYNC 0
#endif

// 16B global -> LDS copy: CDNA5 async path (ASYNCcnt, no VGPR round trip)
// with a synchronous float4 fallback if the builtin is unavailable.
static __device__ __forceinline__ void copy16_g2l(const float* g, float* l) {
#if HAVE_ASYNC
    __builtin_amdgcn_global_load_async_to_lds_b128(
        (GLOBAL_AS v4i*)const_cast<float*>(g), (LDS_AS v4i*)l, 0, 0);
#else
    *(float4*)l = *(const float4*)g;
#endif
}

static __device__ __forceinline__ void wait_g2l() {
#if HAVE_ASYNC
#if __has_builtin(__builtin_amdgcn_s_wait_asynccnt)
    __builtin_amdgcn_s_wait_asynccnt(0);
#else
    asm volatile("s_wait_asynccnt 0x0" ::: "memory");
#endif
#endif
}

struct KDesc {
    int Ku;         // number of real union keys (<= 129)
    int idx[KP];    // gather row per union slot (pad: 0)
    int fd[KP];     // first depth containing this key (pad: NDEPTH)
};

static __device__ __forceinline__ v8f wmma_f32(v2f a, v2f b, v8f c) {
    // D = A(16x4 f32) * B(4x16 f32) + C(16x16 f32), exact fp32
    return __builtin_amdgcn_wmma_f32_16x16x4_f32(false, a, false, b,
                                                 (short)0, c, false, false);
}

__global__ __launch_bounds__(WAVES * 32)
void cantor_attn_kernel(const float* __restrict__ q,
                        const float* __restrict__ kmat,
                        const float* __restrict__ vmat,
                        const float* __restrict__ sw,
                        const float* __restrict__ stemp,
                        float* __restrict__ out,
                        KDesc kd)
{
    extern __shared__ float smem[];
    float* sK   = smem;                 // phase A: K rows; phase C: V rows (row-major)
    float* sQ   = smem + REG1;          // phase A: Q-stage; later: per-wave c-matrix
    int*   gIdx = (int*)(smem + REG1 + REG2);
    int*   gFd  = gIdx + KP;

    const int tid  = threadIdx.x;
    const int lane = tid & 31;
    const int w    = tid >> 5;
    const int b    = blockIdx.y;
    const int q0   = blockIdx.x * BQ;

    for (int i = tid; i < KP; i += WAVES * 32) { gIdx[i] = kd.idx[i]; gFd[i] = kd.fd[i]; }
    __syncthreads();

    const float* Qb = q    + (size_t)b * L_ * D_;
    const float* Kb = kmat + (size_t)b * L_ * D_;
    const float* Vb = vmat + (size_t)b * L_ * D_;
    float*       Ob = out  + (size_t)b * L_ * D_;

    const int m  = lane & 15;          // row/col within 16x16 tile
    const int kh = (lane >> 4) << 1;   // 0 or 2: K-pair selector (WMMA f32 A/B layout)

    // ---------------- Phase A: scores[16 x KP] = Q . K_union^T ----------------
    v8f acc[NT];
    #pragma unroll
    for (int t = 0; t < NT; ++t)
        #pragma unroll
        for (int r = 0; r < 8; ++r) acc[t][r] = 0.f;

    for (int c = 0; c < NCHUNK; ++c) {
        const int d0 = c * DC;
        __syncthreads();
        // stage Q chunk: BQ x DC (8 x 16B per thread)
        for (int i = tid; i < BQ * (DC / 4); i += WAVES * 32) {
            const int row = i >> 4, c4 = i & 15;
            copy16_g2l(Qb + (size_t)(q0 + row) * D_ + d0 + c4 * 4,
                       &sQ[row * QP + c4 * 4]);
        }
        // stage gathered K rows: KP x DC (36 x 16B per thread)
        for (int i = tid; i < KP * (DC / 4); i += WAVES * 32) {
            const int j = i >> 4, c4 = i & 15;
            copy16_g2l(Kb + (size_t)gIdx[j] * D_ + d0 + c4 * 4,
                       &sK[j * KPITCH + c4 * 4]);
        }
        wait_g2l();
        __syncthreads();

        const int mrow = (w << 4) + m;
        #pragma unroll 4
        for (int ks = 0; ks < DC / 4; ++ks) {
            const int k0 = ks * 4 + kh;
            const v2f a = *(const v2f*)&sQ[mrow * QP + k0];
            #pragma unroll
            for (int t = 0; t < NT; ++t) {
                const v2f bb = *(const v2f*)&sK[(t * 16 + m) * KPITCH + k0];
                acc[t] = wmma_f32(a, bb, acc[t]);
            }
        }
    }
    __syncthreads();   // all waves done reading sQ before it becomes sC

    // dump scores (C-layout -> [m][n] in LDS), apply 1/sqrt(D)
    float* sCw = (smem + REG1) + (size_t)w * 16 * CP;
    {
        const int half = (lane >> 4) << 3;
        #pragma unroll
        for (int t = 0; t < NT; ++t)
            #pragma unroll
            for (int r = 0; r < 8; ++r)
                sCw[(half + r) * CP + t * 16 + m] = acc[t][r] * 0.03125f;
    }

    // ---------------- fused multi-depth softmax -> coefficients c_j ----------------
    if (lane < 16) {
        float* row = sCw + lane * CP;
        const int Ku = kd.Ku;
        float mx = -3.0e38f;
        for (int j = 0; j < Ku; ++j) mx = fmaxf(mx, row[j]);

        float Z[NDEPTH];
        #pragma unroll
        for (int d = 0; d < NDEPTH; ++d) Z[d] = 0.f;
        for (int j = 0; j < Ku; ++j) {
            const float e = __expf(row[j] - mx);
            row[j] = e;
            const int f = gFd[j];
            #pragma unroll
            for (int d = 0; d < NDEPTH; ++d) Z[d] += (f <= d) ? e : 0.f;  // nested sets
        }
        // depth weights: softmax(scale_weights / scale_temperature)
        const float st = stemp[0];
        float x[NDEPTH]; float xm = -3.0e38f;
        #pragma unroll
        for (int d = 0; d < NDEPTH; ++d) { x[d] = sw[d] / st; xm = fmaxf(xm, x[d]); }
        float se = 0.f;
        #pragma unroll
        for (int d = 0; d < NDEPTH; ++d) { x[d] = __expf(x[d] - xm); se += x[d]; }
        // suffix sums T_d = sum_{d'>=d} w_d' / Z_d'
        float T[NDEPTH]; float ta = 0.f;
        #pragma unroll
        for (int d = NDEPTH - 1; d >= 0; --d) { ta += (x[d] / se) / Z[d]; T[d] = ta; }
        // c_j = e_j * T[first_depth(j)]
        for (int j = 0; j < Ku; ++j) {
            const int f = gFd[j];
            float ts = 0.f;
            #pragma unroll
            for (int d = 0; d < NDEPTH; ++d) ts = (f == d) ? T[d] : ts;
            row[j] *= ts;
        }
        for (int j = Ku; j < KP; ++j) row[j] = 0.f;   // zero padded columns
    }

    // ---------------- Phase C: out[16 x D] = c . V_union ----------------
    float* sV = sK;  // region1 reused: V rows, row-major [KP][KPITCH]
    for (int c = 0; c < NCHUNK; ++c) {
        const int d0 = c * DC;
        __syncthreads();
        for (int i = tid; i < KP * (DC / 4); i += WAVES * 32) {
            const int j = i >> 4, c4 = i & 15;
            copy16_g2l(Vb + (size_t)gIdx[j] * D_ + d0 + c4 * 4,
                       &sV[j * KPITCH + c4 * 4]);
        }
        wait_g2l();
        __syncthreads();

        #pragma unroll
        for (int dt = 0; dt < DC / 16; ++dt) {
            v8f o;
            #pragma unroll
            for (int r = 0; r < 8; ++r) o[r] = 0.f;
            const int dl = dt * 16 + m;     // local d column for this lane
            #pragma unroll 4
            for (int ks = 0; ks < KP / 4; ++ks) {
                const int k0 = ks * 4 + kh;
                const v2f a = *(const v2f*)&sCw[m * CP + k0];
                // B[kk][n] = V_union[k0+kk][d]; rows 272B apart -> ds_load_2addr_b32
                v2f bb = { sV[(k0 + 0) * KPITCH + dl],
                           sV[(k0 + 1) * KPITCH + dl] };
                o = wmma_f32(a, bb, o);
            }
            const int half = (lane >> 4) << 3;
            const int dg   = d0 + dl;
            #pragma unroll
            for (int r = 0; r < 8; ++r)
                Ob[(size_t)(q0 + (w << 4) + half + r) * D_ + dg] = o[r];
        }
    }
}

// ---------------- host: exact replica of _cantor_indices + nesting ----------------
static KDesc build_desc() {
    KDesc kd;
    std::memset(&kd, 0, sizeof(kd));
    std::vector<std::set<long long>> sets(NDEPTH);
    for (int depth = 0; depth < NDEPTH; ++depth) {
        std::vector<double> pos = {0.0, 1.0};
        for (int it = 0; it < depth; ++it) {
            std::vector<double> nw;
            nw.reserve(pos.size() * 2);
            for (size_t i = 0; i + 1 < pos.size(); ++i) {
                const double l = pos[i], r = pos[i + 1];
                nw.push_back(l);
                nw.push_back(l + (r - l) / 3.0);
            }
            nw.push_back(pos.back());
            pos.swap(nw);
        }
        for (double p : pos) {
            const float pf  = (float)p;                    // .astype(float32)
            const float val = pf * (float)(L_ - 1);        // f32 multiply
            sets[depth].insert((long long)val);            // trunc toward zero
        }
    }
    std::set<long long> uni;
    for (auto& s : sets) uni.insert(s.begin(), s.end());
    int n = 0;
    for (long long e : uni) {
        if (n >= KP) break;
        kd.idx[n] = (int)e;
        int f = NDEPTH;
        for (int d = 0; d < NDEPTH; ++d)
            if (sets[d].count(e)) { f = d; break; }
        kd.fd[n] = f;
        ++n;
    }
    kd.Ku = n;
    for (int j = n; j < KP; ++j) { kd.idx[j] = 0; kd.fd[j] = NDEPTH; }
    return kd;
}

extern "C" void kernel_launch(void* const* d_in, const int* in_sizes, int n_in,
                              void* d_out, int out_size, void* d_ws, size_t ws_size,
                              hipStream_t stream) {
    (void)in_sizes; (void)n_in; (void)out_size; (void)d_ws; (void)ws_size;
    const float* q  = (const float*)d_in[0];
    const float* k  = (const float*)d_in[1];
    const float* v  = (const float*)d_in[2];
    // d_in[3] = t : unused by the reference
    const float* sw = (const float*)d_in[4];
    const float* st = (const float*)d_in[5];
    float* out = (float*)d_out;

    const KDesc kd = build_desc();

    dim3 grid(L_ / BQ, B_);
    dim3 block(WAVES * 32);
    const size_t smem_bytes = (size_t)(REG1 + REG2 + 2 * KP) * sizeof(float); // 59,008 B
    cantor_attn_kernel<<<grid, block, smem_bytes, stream>>>(q, k, v, sw, st, out, kd);
}